// P2RLoss_58351425683703
// MI455X (gfx1250) — compile-verified
//
#include <hip/hip_runtime.h>
#include <math.h>

typedef float v2f __attribute__((ext_vector_type(2)));
typedef float v8f __attribute__((ext_vector_type(8)));

namespace {
constexpr int   kB          = 8;
constexpr int   kH          = 192;
constexpr int   kW          = 192;
constexpr int   kN          = 1000;
constexpr int   kHW         = kH * kW;              // 36864 pixels per batch
constexpr int   kNT         = 63;                   // point tiles of 16 -> 1008 padded
constexpr int   kBlocksPerB = 36;
constexpr int   kPixPerBlk  = kHW / kBlocksPerB;    // 1024
constexpr int   kThreads    = 256;                  // 8 wave32
constexpr int   kWaves      = kThreads / 32;
constexpr int   kTilesPerWv = (kPixPerBlk / 16) / kWaves; // 8 pixel-tiles per wave
constexpr float kBig        = 1e30f;
constexpr float kEps        = 1e-8f;
}

struct Accum {
  float    sumP, spp, spt, stt;   // sum(pred), sum(p^2), sum(p*t), sum(t^2)
  unsigned maxP, minD2;           // ordered-uint encodings of nonneg floats
};

__global__ void p2r_init_kernel(Accum* acc) {
  int i = threadIdx.x;
  if (i < kB) {
    acc[i].sumP = 0.0f; acc[i].spp = 0.0f; acc[i].spt = 0.0f; acc[i].stt = 0.0f;
    acc[i].maxP  = 0u;            // +0.0f
    acc[i].minD2 = 0x7F800000u;   // +inf
  }
}

// One block = 1024 contiguous pixels of one batch image. Each wave sweeps
// 16-pixel tiles against all 1008 (padded) points using V_WMMA_F32_16X16X4_F32:
//   A_pix = [-2x, -2y, x^2+y^2, 1],  B_pt = [px, py, 1, px^2+py^2]
//   => D[m][n] = squared distance, 256 pairs per wmma.
// Min-reduction runs in unsigned-int domain (bit order == float order for
// nonneg values) to avoid fcanonicalize traffic around IEEE minnum.
__global__ void __launch_bounds__(kThreads)
p2r_spatial_kernel(const float* __restrict__ dens,
                   const float* __restrict__ pts,
                   Accum* __restrict__ acc) {
  // B operand pre-swizzled into WMMA layout: index [t*32 + lane]
  __shared__ float2   Blds[kNT * 32];          // 16 KB
  __shared__ float    s_sum[4];
  __shared__ unsigned s_maxP, s_minD2;

  const int b   = blockIdx.x / kBlocksPerB;
  const int blk = blockIdx.x % kBlocksPerB;
  const int tid = threadIdx.x;

  // ---- stage points into LDS in B-matrix layout (pad with huge norm) ----
  const float* __restrict__ P = pts + (size_t)b * (kN * 2);
  for (int idx = tid; idx < kNT * 32; idx += kThreads) {
    const int l = idx & 31;
    const int j = (idx >> 5) * 16 + (l & 15);
    float px = 0.0f, py = 0.0f, nrm = kBig;
    if (j < kN) { px = P[2 * j]; py = P[2 * j + 1]; nrm = px * px + py * py; }
    float2 v;
    if (l < 16) { v.x = px;   v.y = py;  }   // rows K=0 (px), K=1 (py)
    else        { v.x = 1.0f; v.y = nrm; }   // rows K=2 (1),  K=3 (|p|^2)
    Blds[idx] = v;
  }
  if (tid == 0) {
    s_sum[0] = s_sum[1] = s_sum[2] = s_sum[3] = 0.0f;
    s_maxP = 0u; s_minD2 = 0x7F800000u;
  }
  __syncthreads();

  const int lane = tid & 31;
  const int wave = tid >> 5;
  const int half = lane >> 4;      // 0: lanes 0-15, 1: lanes 16-31
  const int lm   = lane & 15;
  const int blockPixBase = blk * kPixPerBlk;
  const float* __restrict__ predB = dens + (size_t)b * kHW;

  float lsumP = 0.0f, lspp = 0.0f, lspt = 0.0f, lstt = 0.0f;
  float lmaxP = 0.0f;
  unsigned lminD2 = 0x7F800000u;

  for (int tt = 0; tt < kTilesPerWv; ++tt) {
    const int tileBase = blockPixBase + (wave * kTilesPerWv + tt) * 16;

    // A operand: 16x4 f32 layout (lanes 0-15: K=0,1; lanes 16-31: K=2,3)
    const int   p  = tileBase + lm;
    const float xc = (float)(p % kW) * 8.0f + 4.0f;
    const float yc = (float)(p / kW) * 8.0f + 4.0f;
    v2f a;
    a.x = half ? (xc * xc + yc * yc) : (-2.0f * xc);
    a.y = half ? 1.0f                : (-2.0f * yc);

    unsigned dmin[8];
#pragma unroll
    for (int k = 0; k < 8; ++k) dmin[k] = 0x7F800000u;   // +inf

    for (int t = 0; t < kNT; ++t) {
      const float2 bv = Blds[t * 32 + lane];     // ds_load_b64
      v2f bb; bb.x = bv.x; bb.y = bv.y;
      v8f c = {};
      // D = A x B + 0 : full 16x16 tile of squared distances
      v8f d = __builtin_amdgcn_wmma_f32_16x16x4_f32(
          false, a, false, bb, (short)0, c, false, false);
#pragma unroll
      for (int k = 0; k < 8; ++k) {
        const unsigned du = __float_as_uint(d[k]);   // nonneg: uint order == float order
        dmin[k] = dmin[k] < du ? dmin[k] : du;       // v_min_u32 / v_min3_u32
      }
    }

    // min over columns (points) = min across lanes within each 16-lane half
#pragma unroll
    for (int k = 0; k < 8; ++k) {
      unsigned m = dmin[k], o;
      o = (unsigned)__shfl_xor((int)m, 1, 32); m = m < o ? m : o;
      o = (unsigned)__shfl_xor((int)m, 2, 32); m = m < o ? m : o;
      o = (unsigned)__shfl_xor((int)m, 4, 32); m = m < o ? m : o;
      o = (unsigned)__shfl_xor((int)m, 8, 32); m = m < o ? m : o;
      dmin[k] = m;
    }

    // writer lanes: lane 0 owns rows M=0..7 (pixels tileBase+0..7),
    //               lane 16 owns rows M=8..15 (pixels tileBase+8..15)
    if (lm == 0) {
      const int rowBase = tileBase + half * 8;
      const float4* pr = (const float4*)(predB + rowBase);  // 32B aligned
      const float4 q0 = pr[0], q1 = pr[1];
      float pv[8] = {q0.x, q0.y, q0.z, q0.w, q1.x, q1.y, q1.z, q1.w};
#pragma unroll
      for (int k = 0; k < 8; ++k) {
        const float md2 = fmaxf(__uint_as_float(dmin[k]), 0.0f);
        const float md  = __builtin_sqrtf(md2);
        // sigmoid((8 - md)/4) = 1/(1+exp((md-8)*0.25))
        const float tg = 1.0f / (1.0f + __expf((md - 8.0f) * 0.25f));
        lsumP += pv[k];
        lspp  += pv[k] * pv[k];
        lspt  += pv[k] * tg;
        lstt  += tg * tg;
        lmaxP  = fmaxf(lmaxP, pv[k]);
        lminD2 = lminD2 < dmin[k] ? lminD2 : dmin[k];
      }
    }
  }

  if (lm == 0) {
    atomicAdd(&s_sum[0], lsumP);
    atomicAdd(&s_sum[1], lspp);
    atomicAdd(&s_sum[2], lspt);
    atomicAdd(&s_sum[3], lstt);
    atomicMax(&s_maxP,  __float_as_uint(lmaxP));
    atomicMin(&s_minD2, lminD2);
  }
  __syncthreads();

  if (tid == 0) {
    atomicAdd(&acc[b].sumP, s_sum[0]);
    atomicAdd(&acc[b].spp,  s_sum[1]);
    atomicAdd(&acc[b].spt,  s_sum[2]);
    atomicAdd(&acc[b].stt,  s_sum[3]);
    atomicMax(&acc[b].maxP,  s_maxP);
    atomicMin(&acc[b].minD2, s_minD2);
  }
}

__global__ void p2r_final_kernel(const Accum* __restrict__ acc,
                                 float* __restrict__ out) {
  if (threadIdx.x == 0 && blockIdx.x == 0) {
    float cl = 0.0f, sl = 0.0f, sp = 0.0f;
    const float invHW = 1.0f / (float)kHW;
    for (int b = 0; b < kB; ++b) {
      const float pc = acc[b].sumP * (1.0f / 64.0f);   // cell_area = 64
      const float c  = fabsf(pc - (float)kN);
      cl += c;
      sl += c * (1.0f / (float)kN);
      const float pm = __uint_as_float(acc[b].maxP);
      const float md = __builtin_sqrtf(fmaxf(__uint_as_float(acc[b].minD2), 0.0f));
      const float tm = 1.0f / (1.0f + __expf((md - 8.0f) * 0.25f));
      const float pmc = fmaxf(pm, kEps), tmc = fmaxf(tm, kEps);
      const float mse = acc[b].spp * invHW / (pmc * pmc)
                      - 2.0f * acc[b].spt * invHW / (pmc * tmc)
                      + acc[b].stt * invHW / (tmc * tmc);
      if (pm > kEps && tm > kEps) sp += mse;
    }
    const float invB = 1.0f / (float)kB;
    out[0] = 2.0f * cl * invB + 0.5f * sl * invB + 0.1f * sp * invB;
  }
}

extern "C" void kernel_launch(void* const* d_in, const int* in_sizes, int n_in,
                              void* d_out, int out_size, void* d_ws, size_t ws_size,
                              hipStream_t stream) {
  const float* dens = (const float*)d_in[0];   // (B,1,H,W) f32
  const float* pts  = (const float*)d_in[1];   // (B,N,2)  f32
  float* out  = (float*)d_out;
  Accum* acc  = (Accum*)d_ws;                  // 8 * 24 bytes

  p2r_init_kernel<<<1, 32, 0, stream>>>(acc);
  p2r_spatial_kernel<<<kB * kBlocksPerB, kThreads, 0, stream>>>(dens, pts, acc);
  p2r_final_kernel<<<1, 32, 0, stream>>>(acc, out);
}